// VoxelAggregation_67997922230554
// MI455X (gfx1250) — compile-verified
//
#include <hip/hip_runtime.h>

#define THREADS 256
#define RADIX   256

typedef float v2f __attribute__((ext_vector_type(2)));
typedef float v8f __attribute__((ext_vector_type(8)));

// ---------- helpers ----------
// order-preserving f32 <-> u32 transform (for signed-float atomic min/max)
__device__ __forceinline__ unsigned f2ord(float f) {
  unsigned u = __float_as_uint(f);
  return (u & 0x80000000u) ? ~u : (u | 0x80000000u);
}
__device__ __forceinline__ float ord2f(unsigned t) {
  unsigned u = (t & 0x80000000u) ? (t & 0x7FFFFFFFu) : ~t;
  return __uint_as_float(u);
}

// header slots (u32): [0..3] pc_min bits (values >=0 so raw-bit min works)
//                     [4..7] vc max (int)   [12] hmin ord   [13] hmax ord
__global__ void k_init(unsigned* hdr) {
  int t = threadIdx.x;
  if (t < 4)  hdr[t] = 0x7F800000u;       // +inf bits
  if (t >= 4 && t < 8) ((int*)hdr)[t] = 0;
  if (t == 12) hdr[12] = 0xFFFFFFFFu;
  if (t == 13) hdr[13] = 0u;
}

__global__ void k_min4(const float* __restrict__ pb, int n, unsigned* hdr) {
  __shared__ unsigned smin[4];
  if (threadIdx.x < 4) smin[threadIdx.x] = 0x7F800000u;
  __syncthreads();
  int i = blockIdx.x * blockDim.x + threadIdx.x;
  if (i < n) {
    atomicMin(&smin[0], __float_as_uint(pb[i * 4 + 0]));
    atomicMin(&smin[1], __float_as_uint(pb[i * 4 + 1]));
    atomicMin(&smin[2], __float_as_uint(pb[i * 4 + 2]));
    atomicMin(&smin[3], __float_as_uint(pb[i * 4 + 3]));
  }
  __syncthreads();
  if (threadIdx.x < 4) atomicMin(&hdr[threadIdx.x], smin[threadIdx.x]);
}

__global__ void k_vcmax(const float* __restrict__ pb, const float* __restrict__ vs,
                        int n, unsigned* hdr) {
  __shared__ int smax[4];
  if (threadIdx.x < 4) smax[threadIdx.x] = 0;
  __syncthreads();
  int i = blockIdx.x * blockDim.x + threadIdx.x;
  if (i < n) {
#pragma unroll
    for (int c = 0; c < 4; ++c) {
      float m = __uint_as_float(hdr[c]);
      int vc = (int)floorf((pb[i * 4 + c] - m) / vs[c]);
      atomicMax(&smax[c], vc);
    }
  }
  __syncthreads();
  if (threadIdx.x < 4) atomicMax(&((int*)hdr)[4 + threadIdx.x], smax[threadIdx.x]);
}

__global__ void k_hminmax(const float* __restrict__ h, int n, unsigned* hdr) {
  __shared__ unsigned smn, smx;
  if (threadIdx.x == 0) { smn = 0xFFFFFFFFu; smx = 0u; }
  __syncthreads();
  int i = blockIdx.x * blockDim.x + threadIdx.x;
  if (i < n) {
    unsigned t = f2ord(h[i]);
    atomicMin(&smn, t);
    atomicMax(&smx, t);
  }
  __syncthreads();
  if (threadIdx.x == 0) { atomicMin(&hdr[12], smn); atomicMax(&hdr[13], smx); }
}

__global__ void k_makekeys(const float* __restrict__ pb, const float* __restrict__ vs,
                           const unsigned* __restrict__ hdr, unsigned* __restrict__ keys,
                           unsigned* __restrict__ vals, float* __restrict__ out_pc, int n) {
  int i = blockIdx.x * blockDim.x + threadIdx.x;
  if (i >= n) return;
  int vc[4];
#pragma unroll
  for (int c = 0; c < 4; ++c) {
    float m = __uint_as_float(hdr[c]);
    vc[c] = (int)floorf((pb[i * 4 + c] - m) / vs[c]);
    out_pc[i * 4 + c] = (float)vc[c];
  }
  unsigned d1 = (unsigned)(((const int*)hdr)[5] + 1);
  unsigned d2 = (unsigned)(((const int*)hdr)[6] + 1);
  unsigned d3 = (unsigned)(((const int*)hdr)[7] + 1);
  unsigned merge = (unsigned)vc[0];
  merge = merge * d1 + (unsigned)vc[1];
  merge = merge * d2 + (unsigned)vc[2];
  merge = merge * d3 + (unsigned)vc[3];
  keys[i] = merge;
  vals[i] = (unsigned)i;
}

// ---------- radix sort (stable, 8-bit digits, wave32 ballot ranking) ----------
__global__ void k_hist(const unsigned* __restrict__ keys, unsigned* __restrict__ hist,
                       int shift, int n, int nB) {
  __shared__ unsigned bins[RADIX];
  bins[threadIdx.x] = 0;
  __syncthreads();
  int i = blockIdx.x * THREADS + threadIdx.x;
  if (i < n) atomicAdd(&bins[(keys[i] >> shift) & 255u], 1u);
  __syncthreads();
  hist[threadIdx.x * nB + blockIdx.x] = bins[threadIdx.x];  // digit-major
}

__global__ void k_scan_blk(const unsigned* __restrict__ in, unsigned* __restrict__ out,
                           unsigned* __restrict__ bsum, int m) {
  __shared__ unsigned s[THREADS];
  int i = blockIdx.x * THREADS + threadIdx.x;
  unsigned v = (i < m) ? in[i] : 0u;
  s[threadIdx.x] = v;
  __syncthreads();
  for (int off = 1; off < THREADS; off <<= 1) {
    unsigned add = (threadIdx.x >= (unsigned)off) ? s[threadIdx.x - off] : 0u;
    __syncthreads();
    s[threadIdx.x] += add;
    __syncthreads();
  }
  if (i < m) out[i] = s[threadIdx.x];
  if (threadIdx.x == THREADS - 1) bsum[blockIdx.x] = s[threadIdx.x];
}

__global__ void k_scan_sums(unsigned* __restrict__ bsum, int nb) {
  __shared__ unsigned s[THREADS];
  __shared__ unsigned carry;
  if (threadIdx.x == 0) carry = 0;
  __syncthreads();
  for (int base = 0; base < nb; base += THREADS) {
    int i = base + threadIdx.x;
    unsigned v = (i < nb) ? bsum[i] : 0u;
    s[threadIdx.x] = v;
    __syncthreads();
    for (int off = 1; off < THREADS; off <<= 1) {
      unsigned add = (threadIdx.x >= (unsigned)off) ? s[threadIdx.x - off] : 0u;
      __syncthreads();
      s[threadIdx.x] += add;
      __syncthreads();
    }
    if (i < nb) bsum[i] = carry + s[threadIdx.x] - v;  // exclusive
    __syncthreads();
    if (threadIdx.x == 0) carry += s[THREADS - 1];
    __syncthreads();
  }
}

__global__ void k_scan_add(unsigned* __restrict__ out, const unsigned* __restrict__ bsum, int m) {
  int i = blockIdx.x * THREADS + threadIdx.x;
  if (i < m) out[i] += bsum[blockIdx.x];
}

// stable scatter: wave32 ballot ranking + per-wave LDS digit counts
__global__ void k_scatter(const unsigned* __restrict__ kin, const unsigned* __restrict__ vin,
                          unsigned* __restrict__ kout, unsigned* __restrict__ vout,
                          const unsigned* __restrict__ histCnt,
                          const unsigned* __restrict__ histIncl,
                          int shift, int n, int nB) {
  __shared__ unsigned wcnt[(THREADS / 32) * RADIX];
  for (int j = threadIdx.x; j < (THREADS / 32) * RADIX; j += THREADS) wcnt[j] = 0;
  __syncthreads();
  int i = blockIdx.x * THREADS + threadIdx.x;
  int lane = threadIdx.x & 31;
  int wave = threadIdx.x >> 5;
  bool act = (i < n);
  unsigned key = 0, val = 0, d = 0;
  if (act) {
    key = kin[i];
    if (vin) val = vin[i];
    d = (key >> shift) & 255u;
  }
  unsigned m = (unsigned)__ballot(act);  // wave32: low 32 bits
#pragma unroll
  for (int b = 0; b < 8; ++b) {
    unsigned bal = (unsigned)__ballot(act && ((d >> b) & 1u));
    m &= ((d >> b) & 1u) ? bal : ~bal;
  }
  unsigned rank = __popc(m & ((1u << lane) - 1u));
  unsigned cnt = __popc(m);
  if (act && rank == 0) wcnt[wave * RADIX + d] = cnt;
  __syncthreads();
  if (act) {
    unsigned base = 0;
    for (int w = 0; w < wave; ++w) base += wcnt[w * RADIX + d];
    unsigned idx = d * (unsigned)nB + (unsigned)blockIdx.x;
    unsigned gbase = histIncl[idx] - histCnt[idx];  // exclusive global base
    unsigned pos = gbase + base + rank;
    kout[pos] = key;
    if (vout) vout[pos] = val;
  }
}

// ---------- segments ----------
__global__ void k_flags(const unsigned* __restrict__ keys, unsigned* __restrict__ flags, int n) {
  int i = blockIdx.x * blockDim.x + threadIdx.x;
  if (i < n) flags[i] = (i > 0 && keys[i] != keys[i - 1]) ? 1u : 0u;
}

__global__ void k_heads(const unsigned* __restrict__ keys, const unsigned* __restrict__ segid,
                        const unsigned* __restrict__ svals, unsigned* __restrict__ offsets,
                        float* __restrict__ out_vidx, unsigned* __restrict__ vidx_u32, int n) {
  int i = blockIdx.x * blockDim.x + threadIdx.x;
  if (i >= n) return;
  bool head = (i == 0) || (keys[i] != keys[i - 1]);
  if (head) offsets[segid[i]] = (unsigned)i;
  if (i == n - 1) offsets[segid[i] + 1] = (unsigned)n;
  unsigned p = svals[i];
  out_vidx[p] = (float)segid[i];
  vidx_u32[p] = segid[i];
}

__global__ void k_tval(const float* __restrict__ h, const unsigned* __restrict__ vidx,
                       const unsigned* __restrict__ hdr, unsigned* __restrict__ tkeys, int n) {
  int i = blockIdx.x * blockDim.x + threadIdx.x;
  if (i >= n) return;
  float hmin = ord2f(hdr[12]);
  float hmax = ord2f(hdr[13]);
  float t = h[i] - hmin + (float)vidx[i] * hmax;  // >= 0 when hmax >= 0
  tkeys[i] = __float_as_uint(t);
}

__global__ void k_zero(float* __restrict__ p, long long m) {
  long long i = (long long)blockIdx.x * blockDim.x + threadIdx.x;
  if (i < m) p[i] = 0.0f;
}

// ---------- WMMA segmented feature sum ----------
// Each wave handles 16 consecutive sorted points. Within a tile the local
// segment index r = segid[pos]-segid[s0] is in [0,15] (16 points -> <=16
// segments), which exactly fits M=16. For K-chunks of 4 points:
//   A[r][p] = (local_seg(point p)==r) ? 1 : 0   (16x4 one-hot, f32)
//   B[p][f] = feat[point p][half*16+f]          (4x16, f32)
// D += A*B via v_wmma_f32_16x16x4_f32 -> exact f32 segmented sums.
// A layout (ISA 7.12.2): VGPR0 lanes0-15 K=0 / lanes16-31 K=2; VGPR1 K=1/K=3.
// B mirrors (row striped across lanes). C/D: lane l holds N=l%16,
// rows M = vgpr + (l>=16 ? 8 : 0).
__global__ void k_wmma_feat(const unsigned* __restrict__ segid,
                            const unsigned* __restrict__ svals,
                            const float* __restrict__ feat,
                            float* __restrict__ out_feat, int n) {
  const int wpb = THREADS / 32;
  int wid = blockIdx.x * wpb + (threadIdx.x >> 5);
  int lane = threadIdx.x & 31;
  int ntiles = (n + 15) / 16;
  if (wid >= ntiles) return;  // whole-wave uniform exit
  int s0 = wid * 16;
  int myi = s0 + (lane & 15);
  int clampi = (myi < n) ? myi : (n - 1);
  unsigned myseg = segid[clampi];
  unsigned mypt = svals[clampi];
  unsigned segbase = __shfl(myseg, 0);
  unsigned rmax = __shfl(myseg, 15) - segbase;
  int M = lane & 15;
  int k0 = (lane < 16) ? 0 : 2;

  for (int half = 0; half < 2; ++half) {
    v8f acc = {};
#pragma unroll
    for (int c = 0; c < 4; ++c) {
      int j0 = 4 * c + k0;
      unsigned r0 = __shfl(myseg, j0) - segbase;
      unsigned r1 = __shfl(myseg, j0 + 1) - segbase;
      unsigned p0 = __shfl(mypt, j0);
      unsigned p1 = __shfl(mypt, j0 + 1);
      bool ok0 = (s0 + j0) < n;
      bool ok1 = (s0 + j0 + 1) < n;
      v2f a, b;
      a.x = (ok0 && r0 == (unsigned)M) ? 1.0f : 0.0f;
      a.y = (ok1 && r1 == (unsigned)M) ? 1.0f : 0.0f;
      int col = half * 16 + (lane & 15);
      b.x = feat[(long long)p0 * 32 + col];
      b.y = feat[(long long)p1 * 32 + col];
      acc = __builtin_amdgcn_wmma_f32_16x16x4_f32(false, a, false, b, (short)0, acc,
                                                  false, false);
    }
    int colg = half * 16 + (lane & 15);
    int rowoff = (lane >= 16) ? 8 : 0;
#pragma unroll
    for (int v = 0; v < 8; ++v) {
      unsigned row = (unsigned)(v + rowoff);
      if (row <= rmax) {
        float val = acc[v];
        if (val != 0.0f)
          atomicAdd(&out_feat[(long long)(segbase + row) * 32 + colg], val);
      }
    }
  }
}

// ---------- per-voxel outputs ----------
__global__ void k_voxel(const unsigned* __restrict__ keys, const unsigned* __restrict__ svals,
                        const unsigned* __restrict__ offsets, const unsigned* __restrict__ tkeys,
                        const float* __restrict__ pb, const float* __restrict__ vs,
                        const unsigned* __restrict__ hdr,
                        float* __restrict__ out_vbi, float* __restrict__ out_xyz,
                        float* __restrict__ out_center, float* __restrict__ out_bcenter,
                        float* __restrict__ out_unq, float* __restrict__ out_bxyz,
                        float* __restrict__ out_med, int V) {
  int v = blockIdx.x * blockDim.x + threadIdx.x;
  if (v >= V) return;
  unsigned off = offsets[v], end = offsets[v + 1];
  float cntf = (float)(end - off);
  unsigned d1 = (unsigned)(((const int*)hdr)[5] + 1);
  unsigned d2 = (unsigned)(((const int*)hdr)[6] + 1);
  unsigned d3 = (unsigned)(((const int*)hdr)[7] + 1);
  unsigned rem = keys[off];
  unsigned c3 = rem % d3; rem /= d3;
  unsigned c2 = rem % d2; rem /= d2;
  unsigned c1 = rem % d1; rem /= d1;
  unsigned c0 = rem;
  float s0 = 0, s1 = 0, s2 = 0, s3 = 0;
  for (unsigned j = off; j < end; ++j) {
    unsigned p = svals[j];
    s0 += pb[(long long)p * 4 + 0];
    s1 += pb[(long long)p * 4 + 1];
    s2 += pb[(long long)p * 4 + 2];
    s3 += pb[(long long)p * 4 + 3];
  }
  float b0 = s0 / cntf, x = s1 / cntf, y = s2 / cntf, z = s3 / cntf;
  float m1 = __uint_as_float(hdr[1]), m2 = __uint_as_float(hdr[2]), m3 = __uint_as_float(hdr[3]);
  float cx = (float)c1 * vs[1] + m1 + 0.5f * vs[1];
  float cy = (float)c2 * vs[2] + m2 + 0.5f * vs[2];
  float cz = (float)c3 * vs[3] + m3 + 0.5f * vs[3];
  out_vbi[v] = rintf(b0);
  out_xyz[v * 3 + 0] = x; out_xyz[v * 3 + 1] = y; out_xyz[v * 3 + 2] = z;
  out_center[v * 3 + 0] = cx; out_center[v * 3 + 1] = cy; out_center[v * 3 + 2] = cz;
  out_bcenter[v * 4 + 0] = b0; out_bcenter[v * 4 + 1] = cx;
  out_bcenter[v * 4 + 2] = cy; out_bcenter[v * 4 + 3] = cz;
  out_unq[v * 4 + 0] = (float)c0; out_unq[v * 4 + 1] = (float)c1;
  out_unq[v * 4 + 2] = (float)c2; out_unq[v * 4 + 3] = (float)c3;
  out_bxyz[v * 4 + 0] = b0; out_bxyz[v * 4 + 1] = x;
  out_bxyz[v * 4 + 2] = y; out_bxyz[v * 4 + 3] = z;
  unsigned mo = off + (end - off) / 2;
  float hmax = ord2f(hdr[13]);
  out_med[v] = __uint_as_float(tkeys[mo]) - (float)v * hmax;
}

__global__ void k_feat_div(float* __restrict__ out_feat, const unsigned* __restrict__ offsets,
                           long long m) {
  long long i = (long long)blockIdx.x * blockDim.x + threadIdx.x;
  if (i >= m) return;
  int v = (int)(i >> 5);
  float cnt = (float)(offsets[v + 1] - offsets[v]);
  out_feat[i] /= cnt;
}

// ---------- host ----------
extern "C" void kernel_launch(void* const* d_in, const int* in_sizes, int n_in,
                              void* d_out, int out_size, void* d_ws, size_t ws_size,
                              hipStream_t stream) {
  const float* pb = (const float*)d_in[0];  // point_bxyz  (N,4)
  const float* pf = (const float*)d_in[1];  // point_feat  (N,32)
  const float* ph = (const float*)d_in[2];  // point_height(N,)
  const float* vs = (const float*)d_in[3];  // voxel_size  (4,)
  const int N = in_sizes[0] / 4;
  const long long V = ((long long)out_size - 5LL * N) / 52LL;
  if (N <= 0 || V <= 0) return;

  float* out = (float*)d_out;
  float* out_vbi     = out;
  float* out_xyz     = out + V;
  float* out_center  = out + 4 * V;
  float* out_bcenter = out + 7 * V;
  float* out_unq     = out + 11 * V;
  float* out_bxyz    = out + 15 * V;
  float* out_feat    = out + 19 * V;
  float* out_med     = out + 51 * V;
  float* out_vidx    = out + 52 * V;
  float* out_pc      = out + 52 * V + N;

  // carve workspace
  char* w = (char*)d_ws;
  auto carve = [&](size_t bytes) -> void* {
    void* p = (void*)w;
    w += (bytes + 255) & ~(size_t)255;
    return p;
  };
  const int nPB = (N + THREADS - 1) / THREADS;      // point blocks == radix blocks
  const int histLen = RADIX * nPB;
  unsigned* hdr      = (unsigned*)carve(256);
  unsigned* keysA    = (unsigned*)carve((size_t)N * 4);
  unsigned* keysB    = (unsigned*)carve((size_t)N * 4);
  unsigned* valsA    = (unsigned*)carve((size_t)N * 4);
  unsigned* valsB    = (unsigned*)carve((size_t)N * 4);
  unsigned* tkA      = (unsigned*)carve((size_t)N * 4);
  unsigned* tkB      = (unsigned*)carve((size_t)N * 4);
  unsigned* flags    = (unsigned*)carve((size_t)N * 4);
  unsigned* segid    = (unsigned*)carve((size_t)N * 4);
  unsigned* vidx_u32 = (unsigned*)carve((size_t)N * 4);
  unsigned* offsets  = (unsigned*)carve(((size_t)N + 1) * 4);
  unsigned* histA    = (unsigned*)carve((size_t)histLen * 4);
  unsigned* histB    = (unsigned*)carve((size_t)histLen * 4);
  unsigned* bsum     = (unsigned*)carve((size_t)(nPB + 1) * 4);
  (void)ws_size;

  auto sortPass = [&](const unsigned* kin, const unsigned* vin, unsigned* kout,
                      unsigned* vout, int shift) {
    k_hist<<<nPB, THREADS, 0, stream>>>(kin, histA, shift, N, nPB);
    k_scan_blk<<<nPB, THREADS, 0, stream>>>(histA, histB, bsum, histLen);
    k_scan_sums<<<1, THREADS, 0, stream>>>(bsum, nPB);
    k_scan_add<<<nPB, THREADS, 0, stream>>>(histB, bsum, histLen);
    k_scatter<<<nPB, THREADS, 0, stream>>>(kin, vin, kout, vout, histA, histB, shift, N, nPB);
  };

  // 1) reductions
  k_init<<<1, 32, 0, stream>>>(hdr);
  k_min4<<<nPB, THREADS, 0, stream>>>(pb, N, hdr);
  k_vcmax<<<nPB, THREADS, 0, stream>>>(pb, vs, N, hdr);
  k_hminmax<<<nPB, THREADS, 0, stream>>>(ph, N, hdr);

  // 2) keys + point_coords
  k_makekeys<<<nPB, THREADS, 0, stream>>>(pb, vs, hdr, keysA, valsA, out_pc, N);

  // 3) sort (key, point-idx) — 4 stable passes, ends in keysA/valsA
  sortPass(keysA, valsA, keysB, valsB, 0);
  sortPass(keysB, valsB, keysA, valsA, 8);
  sortPass(keysA, valsA, keysB, valsB, 16);
  sortPass(keysB, valsB, keysA, valsA, 24);

  // 4) segments: flags -> inclusive scan -> segid; heads -> offsets, voxel_index
  k_flags<<<nPB, THREADS, 0, stream>>>(keysA, flags, N);
  k_scan_blk<<<nPB, THREADS, 0, stream>>>(flags, segid, bsum, N);
  k_scan_sums<<<1, THREADS, 0, stream>>>(bsum, nPB);
  k_scan_add<<<nPB, THREADS, 0, stream>>>(segid, bsum, N);
  k_heads<<<nPB, THREADS, 0, stream>>>(keysA, segid, valsA, offsets, out_vidx, vidx_u32, N);

  // 5) median machinery: tval, global sort of 2M floats (value-only)
  k_tval<<<nPB, THREADS, 0, stream>>>(ph, vidx_u32, hdr, tkA, N);
  sortPass(tkA, nullptr, tkB, nullptr, 0);
  sortPass(tkB, nullptr, tkA, nullptr, 8);
  sortPass(tkA, nullptr, tkB, nullptr, 16);
  sortPass(tkB, nullptr, tkA, nullptr, 24);

  // 6) feature segment-sum via V_WMMA_F32_16X16X4_F32, then divide by counts
  long long featElems = V * 32;
  int featBlocks = (int)((featElems + THREADS - 1) / THREADS);
  k_zero<<<featBlocks, THREADS, 0, stream>>>(out_feat, featElems);
  int ntiles = (N + 15) / 16;
  int wmmaBlocks = (ntiles + (THREADS / 32) - 1) / (THREADS / 32);
  k_wmma_feat<<<wmmaBlocks, THREADS, 0, stream>>>(segid, valsA, pf, out_feat, N);

  // 7) per-voxel outputs
  int vBlocks = (int)((V + THREADS - 1) / THREADS);
  k_voxel<<<vBlocks, THREADS, 0, stream>>>(keysA, valsA, offsets, tkA, pb, vs, hdr,
                                           out_vbi, out_xyz, out_center, out_bcenter,
                                           out_unq, out_bxyz, out_med, (int)V);
  k_feat_div<<<featBlocks, THREADS, 0, stream>>>(out_feat, offsets, featElems);
}